// GaussianMixture_46050639347961
// MI455X (gfx1250) — compile-verified
//
#include <hip/hip_runtime.h>
#include <hip/hip_bf16.h>
#include <math.h>

typedef __attribute__((ext_vector_type(2))) float v2f;
typedef __attribute__((ext_vector_type(8))) float v8f;

#define M_COMP 1024
#define N_SAMP 65536
#define NWAVES (N_SAMP / 16)          // 4096 waves, 16 sample-rows each
#define B_FLOATS (8 * M_COMP)         // 4*M float2 pairs = 32 KB
#define PART_OFF (B_FLOATS + 16)      // float offset of per-wave partials in ws
#define LOG2E 1.4426950408889634f
#define LN2   0.6931471805599453f

// ---------------------------------------------------------------------------
// Kernel 1: build the pre-scaled coefficient matrix and logsumexp shift.
//   score[n,m]       = C0 + C1*x + C2*y + C3*x^2 + C4*x*y + C5*y^2 = wlog-qf
//   stored (log2 domain, shift folded into C0):  C'k = log2e*Ck,
//   C'0 = log2e*(C0 - shift)  so WMMA yields log2(exp(score-shift)) directly.
// Layout (float2 pairs, half-wave indexed, K=6,7 zero-padded):
//   B[0*M + m] = {C'0, C'1}   (K=0,1  -> lanes 0-15)
//   B[1*M + m] = {C'2, C'3}   (K=2,3  -> lanes 16-31)
//   B[2*M + m] = {C'4, C'5}   (K=4,5  -> lanes 0-15)
//   B[3*M + m] = {0,   0  }   (K=6,7  -> lanes 16-31)
//   ws[8*M] = shift
// ---------------------------------------------------------------------------
__global__ void __launch_bounds__(M_COMP)
gmm_prep(const float* __restrict__ mu,
         const float* __restrict__ sigma_log,
         const float* __restrict__ theta,
         const float* __restrict__ w,
         float* __restrict__ ws) {
    __shared__ float red[M_COMP];
    const int m = threadIdx.x;

    // log_softmax(w): wmax, then log(sum(exp(w - wmax)))
    float wm = w[m];
    red[m] = wm;
    __syncthreads();
    for (int s = M_COMP / 2; s > 0; s >>= 1) {
        if (m < s) red[m] = fmaxf(red[m], red[m + s]);
        __syncthreads();
    }
    float wmaxv = red[0];
    __syncthreads();
    red[m] = __expf(wm - wmaxv);
    __syncthreads();
    for (int s = M_COMP / 2; s > 0; s >>= 1) {
        if (m < s) red[m] += red[m + s];
        __syncthreads();
    }
    float lz = wmaxv + __logf(red[0]);   // logsumexp(w)
    __syncthreads();

    float sl0 = sigma_log[2 * m], sl1 = sigma_log[2 * m + 1];
    float wlog = wm - lz - (sl0 + sl1);

    // shift = max_m wlog[m]  (upper bound of all scores since qf >= 0)
    red[m] = wlog;
    __syncthreads();
    for (int s = M_COMP / 2; s > 0; s >>= 1) {
        if (m < s) red[m] = fmaxf(red[m], red[m + s]);
        __syncthreads();
    }
    float shift = red[0];

    float a = __expf(-2.0f * sl0);
    float b = __expf(-2.0f * sl1);
    float st, ct;
    __sincosf(theta[m], &st, &ct);
    float g11 = a * ct * ct + b * st * st;
    float g12 = (a - b) * ct * st;
    float g22 = a * st * st + b * ct * ct;

    float mx = mu[2 * m], my = mu[2 * m + 1];
    float c0 = wlog - (g11 * mx * mx + 2.0f * g12 * mx * my + g22 * my * my);
    float c1 = 2.0f * (g11 * mx + g12 * my);
    float c2 = 2.0f * (g12 * mx + g22 * my);
    float c3 = -g11;
    float c4 = -2.0f * g12;
    float c5 = -g22;

    v2f* B = (v2f*)ws;
    B[0 * M_COMP + m] = (v2f){LOG2E * (c0 - shift), LOG2E * c1};
    B[1 * M_COMP + m] = (v2f){LOG2E * c2,           LOG2E * c3};
    B[2 * M_COMP + m] = (v2f){LOG2E * c4,           LOG2E * c5};
    B[3 * M_COMP + m] = (v2f){0.0f, 0.0f};
    if (m == 0) ws[B_FLOATS] = shift;
}

// ---------------------------------------------------------------------------
// Kernel 2: score GEMM via V_WMMA_F32_16X16X4_F32 + sum of exp2.
// One wave handles 16 sample rows vs all 1024 components (64 col tiles).
// A (16x4 f32): lanes 0-15 hold K=0,1; lanes 16-31 hold K=2,3 (vgpr0,1).
// B (4x16 f32): mirrored. C/D (16x16 f32): row = vgpr + 8*(lane>=16), col = lane%16.
// Branch-free B fetch: two ds_load_b64 per tile via half-wave-indexed layout.
// ---------------------------------------------------------------------------
__global__ void __launch_bounds__(256)
gmm_main(const float* __restrict__ sample,
         const float* __restrict__ ws,
         float* __restrict__ partials) {
    __shared__ v2f Bs[4 * M_COMP];     // 32 KB coefficient stage per block
    {
        const v2f* Bg = (const v2f*)ws;
        for (int i = threadIdx.x; i < 4 * M_COMP; i += 256)
            Bs[i] = Bg[i];
    }
    __syncthreads();

    const float shift = ws[B_FLOATS];

    const int lane = threadIdx.x & 31;
    const int wid  = threadIdx.x >> 5;
    const int gw   = blockIdx.x * 8 + wid;   // global wave id, 0..4095
    const int r    = lane & 15;              // row/col within tile
    const int hf   = lane >> 4;              // half-wave id
    const int row  = gw * 16 + r;            // sample row

    const float x = sample[2 * row];
    const float y = sample[2 * row + 1];

    // Feature vector F = [1, x, y, x^2, x*y, y^2, 0, 0] split K=0..3 / K=4..7
    v2f a_lo, a_hi;
    if (hf == 0) { a_lo = (v2f){1.0f, x};      a_hi = (v2f){x * y, y * y}; }
    else         { a_lo = (v2f){y,    x * x};  a_hi = (v2f){0.0f, 0.0f};   }

    const v2f* Blo = Bs + hf * M_COMP;             // K=0,1 or K=2,3
    const v2f* Bhi = Bs + 2 * M_COMP + hf * M_COMP; // K=4,5 or zeros

    float sums[8];
#pragma unroll
    for (int i = 0; i < 8; ++i) sums[i] = 0.0f;

#pragma unroll 4
    for (int t = 0; t < 64; ++t) {
        const int col = t * 16 + r;
        v2f b_lo = Blo[col];
        v2f b_hi = Bhi[col];

        v8f acc = {};
        acc = __builtin_amdgcn_wmma_f32_16x16x4_f32(false, a_lo, false, b_lo,
                                                    (short)0, acc, false, false);
        acc = __builtin_amdgcn_wmma_f32_16x16x4_f32(false, a_hi, false, b_hi,
                                                    (short)0, acc, false, false);

        // acc[i] = log2(exp(score - shift)); accumulate 2^acc
#pragma unroll
        for (int i = 0; i < 8; ++i)
            sums[i] += __builtin_amdgcn_exp2f(acc[i]);
    }

    // Reduce each row's partial sums across the 16 lanes of its half-wave.
#pragma unroll
    for (int i = 0; i < 8; ++i) {
#pragma unroll
        for (int msk = 8; msk >= 1; msk >>= 1)
            sums[i] += __shfl_xor(sums[i], msk, 16);
    }

    // ll[row] = shift + ln2 * log2(rowsum); sum this wave's 16 rows.
    float p = 0.0f;
#pragma unroll
    for (int i = 0; i < 8; ++i) p += __builtin_amdgcn_logf(sums[i]);
    p += __shfl_xor(p, 16, 32);          // combine the two half-waves
    if (lane == 0) partials[gw] = 16.0f * shift + LN2 * p;
}

// ---------------------------------------------------------------------------
// Kernel 3: deterministic fixed-order reduction of the 4096 wave partials.
// ---------------------------------------------------------------------------
__global__ void __launch_bounds__(256)
gmm_final(const float* __restrict__ partials, float* __restrict__ out) {
    __shared__ float red[256];
    const int tid = threadIdx.x;
    float s = 0.0f;
    for (int j = 0; j < NWAVES / 256; ++j)
        s += partials[j * 256 + tid];
    red[tid] = s;
    __syncthreads();
    for (int st = 128; st > 0; st >>= 1) {
        if (tid < st) red[tid] += red[tid + st];
        __syncthreads();
    }
    if (tid == 0) out[0] = -red[0];   // nll = -sum(ll); SPARSITY == 0
}

extern "C" void kernel_launch(void* const* d_in, const int* in_sizes, int n_in,
                              void* d_out, int out_size, void* d_ws, size_t ws_size,
                              hipStream_t stream) {
    const float* sample    = (const float*)d_in[0];
    const float* mu        = (const float*)d_in[1];
    const float* sigma_log = (const float*)d_in[2];
    const float* theta     = (const float*)d_in[3];
    const float* w         = (const float*)d_in[4];
    float* ws = (float*)d_ws;

    gmm_prep<<<1, M_COMP, 0, stream>>>(mu, sigma_log, theta, w, ws);
    gmm_main<<<N_SAMP / 128, 256, 0, stream>>>(sample, ws, ws + PART_OFF);
    gmm_final<<<1, 256, 0, stream>>>(ws + PART_OFF, (float*)d_out);
}